// BucketedGoWatti_79886391705811
// MI455X (gfx1250) — compile-verified
//
#include <hip/hip_runtime.h>
#include <hip/hip_bf16.h>
#include <math.h>

// ---------------------------------------------------------------------------
// BucketedGoWatti for MI455X (gfx1250, wave32).
// All GEMMs use V_WMMA_F32_16X16X4_F32 (exact fp32 path).
//
// Problem constants (from reference):
//   B=4, L=8192, d_h=1024, T=512, d_g=256, D_PROJ=256, WIN=1536, STRIDE=384
//   windows: s_w = min(384*w, L-WIN) for w in [0,19)   (dynamic_slice clamp)
//
// Math rewrite: lw[t] = qt[t,:] . (Zk[t,:] @ Wk_win) = Zk[t,:] . u[t,:]
//   where u = G @ (Wq_win @ Wk_win^T)  -> removes per-window 1024x1024 GEMM.
//
// Occupancy: the 19-window online-softmax loop is split into 2 window groups
// per (b, t-tile) -> 256 workgroups instead of 128; partials are merged with
// an exact flash-combine pass (each window still processed exactly once).
// ---------------------------------------------------------------------------

typedef float v2f __attribute__((ext_vector_type(2)));
typedef float v8f __attribute__((ext_vector_type(8)));

#define B_    4
#define L_    8192
#define DH_   1024
#define T_    512
#define DG_   256
#define P_    256
#define WIN_  1536
#define NW_   19
#define NG_   2                    // window groups (grid parallelism)
#define SMAX_ (L_ - WIN_)          // 6656
#define APAD_ 1540                 // 1536 + 4 : stride%64==4 -> conflict-free A-frag reads
#define RT_   (B_ * T_)            // 2048 query rows

__device__ __forceinline__ v8f wmma_f32_4(v2f a, v2f b, v8f c) {
  // (neg_a, A, neg_b, B, c_mod, C, reuse_a, reuse_b)
  return __builtin_amdgcn_wmma_f32_16x16x4_f32(false, a, false, b, (short)0, c,
                                               false, false);
}

__device__ __forceinline__ float wave_max32(float v) {
#pragma unroll
  for (int off = 16; off > 0; off >>= 1) v = fmaxf(v, __shfl_xor(v, off, 32));
  return v;
}
__device__ __forceinline__ float wave_sum32(float v) {
#pragma unroll
  for (int off = 16; off > 0; off >>= 1) v += __shfl_xor(v, off, 32);
  return v;
}

// ---------------------------------------------------------------------------
// Generic fp32 WMMA GEMM:  C[R,N] = A[R,K] @ B   (B is [K,N], or [N,K] if TB)
// block = 256 threads = 8 waves arranged 2x4; block tile = 32 rows x 256 cols
// wave tile = 16 rows x 64 cols (4 C-tiles of 16x16)
// ---------------------------------------------------------------------------
template <bool TB>
__global__ void gemm_wmma_f32(const float* __restrict__ A,
                              const float* __restrict__ Bm,
                              float* __restrict__ C, int R, int K, int N) {
  const int lane = threadIdx.x & 31;
  const int wv   = threadIdx.x >> 5;
  const int r    = lane & 15;
  const int kh   = lane >> 4;                    // 0: K0/K1 half, 1: K2/K3 half
  const int row0 = blockIdx.x * 32 + (wv >> 2) * 16;
  const int col0 = blockIdx.y * 256 + (wv & 3) * 64;

  v8f acc[4] = {};
  for (int k0 = 0; k0 < K; k0 += 4) {
    const int ka = k0 + 2 * kh;
    v2f a;
    const float* ap = A + (size_t)(row0 + r) * K + ka;
    a.x = ap[0];
    a.y = ap[1];
#pragma unroll
    for (int ct = 0; ct < 4; ++ct) {
      const int col = col0 + ct * 16 + r;
      v2f b;
      if (TB) {
        const float* bp = Bm + (size_t)col * K + ka;
        b.x = bp[0];
        b.y = bp[1];
      } else {
        const float* bp = Bm + (size_t)ka * N + col;
        b.x = bp[0];
        b.y = bp[N];
      }
      acc[ct] = wmma_f32_4(a, b, acc[ct]);
    }
  }
#pragma unroll
  for (int ct = 0; ct < 4; ++ct) {
    const int col = col0 + ct * 16 + r;
#pragma unroll
    for (int j = 0; j < 8; ++j) {
      const int row = row0 + j + 8 * kh;
      C[(size_t)row * N + col] = acc[ct][j];
    }
  }
}

// ---------------------------------------------------------------------------
// K projection with per-batch transposed store:
//   KcT[b][p][l] = (H @ Wk_core)[b*L + l][p]
// so the scores-GEMM B-fragment (needs Kc^T) loads coalesced along l.
// R = B*L = 32768, K = 1024, N = 256
// ---------------------------------------------------------------------------
__global__ void kproj_wmma_f32(const float* __restrict__ H,
                               const float* __restrict__ Wk,
                               float* __restrict__ KcT) {
  const int lane = threadIdx.x & 31;
  const int wv   = threadIdx.x >> 5;
  const int r    = lane & 15;
  const int kh   = lane >> 4;
  const int row0 = blockIdx.x * 32 + (wv >> 2) * 16;   // over B*L
  const int col0 = (wv & 3) * 64;                      // over P (256)

  v8f acc[4] = {};
  for (int k0 = 0; k0 < DH_; k0 += 4) {
    const int ka = k0 + 2 * kh;
    v2f a;
    const float* ap = H + (size_t)(row0 + r) * DH_ + ka;
    a.x = ap[0];
    a.y = ap[1];
#pragma unroll
    for (int ct = 0; ct < 4; ++ct) {
      const int col = col0 + ct * 16 + r;
      v2f b;
      const float* bp = Wk + (size_t)ka * P_ + col;
      b.x = bp[0];
      b.y = bp[P_];
      acc[ct] = wmma_f32_4(a, b, acc[ct]);
    }
  }
#pragma unroll
  for (int ct = 0; ct < 4; ++ct) {
    const int p = col0 + ct * 16 + r;
#pragma unroll
    for (int j = 0; j < 8; ++j) {
      const int row = row0 + j + 8 * kh;          // global residue row
      const int b   = row >> 13;                  // /L_
      const int l   = row & (L_ - 1);
      KcT[((size_t)b * P_ + p) * L_ + l] = acc[ct][j];
    }
  }
}

// ---------------------------------------------------------------------------
// Fused windowed flash-pool kernel (one window group per block).
// One block (256 thr = 8 wave32) owns (b, 16-row T-tile, window group);
// loops its windows with the exact online-softmax recurrence:
//   scores[16,1536] = qc_tile @ Kc_win^T * P^-0.5   (12 WMMA tiles / wave)
//   softmax row-wise in LDS (masked)
//   Zk[16,1024]    = alpha @ H_win                  (8 WMMA tiles / wave)
//   lw[t] = D^-0.5 * sum_e Zk[t,e]*u[t,e]           (deterministic LDS tree)
//   online-softmax update of (m, ssum, z) ; z register-resident (C-layout)
// Emits unnormalized partial (m, ssum, z) for the group.
// ---------------------------------------------------------------------------
__launch_bounds__(256, 1)
__global__ void fused_gowatti(const float* __restrict__ qc,   // [RT, 256]
                              const float* __restrict__ u,    // [RT, 1024]
                              const float* __restrict__ KcT,  // [B, 256, L]
                              const float* __restrict__ H,    // [B, L, 1024]
                              const unsigned char* __restrict__ mask, // [B,L]
                              float* __restrict__ zpart,      // [NG, RT, 1024]
                              float* __restrict__ mpart,      // [NG, RT]
                              float* __restrict__ spart) {    // [NG, RT]
  __shared__ float s_alpha[16 * APAD_];   // scores -> alpha (98.5 KB)
  __shared__ float s_part[16][128];       // lw partials (8 KB)
  __shared__ float s_m[16], s_sum[16], s_em[16], s_ew[16];

  const int tid  = threadIdx.x;
  const int lane = tid & 31;
  const int wv   = tid >> 5;
  const int r    = lane & 15;
  const int kh   = lane >> 4;

  const int b    = blockIdx.x >> 5;        // 32 T-tiles per batch
  const int t0   = (blockIdx.x & 31) * 16;
  const int g    = blockIdx.y;             // window group
  const int wbeg = g * 10;                 // group0: [0,10)  group1: [10,19)
  const int wend = min(wbeg + 10, NW_);
  const int rowQ = b * T_ + t0;            // base row into qc/u/outputs

  if (tid < 16) { s_m[tid] = -INFINITY; s_sum[tid] = 0.f; }

  v8f z[8];                                 // running numerator, wave cols
#pragma unroll
  for (int ct = 0; ct < 8; ++ct) z[ct] = (v8f){0.f,0.f,0.f,0.f,0.f,0.f,0.f,0.f};
  const int col0w = wv * 128;               // this wave's 128 d_h columns

  __syncthreads();

  for (int w = wbeg; w < wend; ++w) {
    const int s = min(384 * w, SMAX_);

    // ---- stage 1: scores = qc_tile @ Kc_win^T, 12 col-tiles per wave ----
    {
      v8f sa[12];
#pragma unroll
      for (int i = 0; i < 12; ++i) sa[i] = (v8f){0.f,0.f,0.f,0.f,0.f,0.f,0.f,0.f};
      const int cbase = wv * 192;           // window-local col base
      for (int k0 = 0; k0 < P_; k0 += 4) {
        const int ka = k0 + 2 * kh;
        v2f a;
        const float* ap = qc + (size_t)(rowQ + r) * P_ + ka;
        a.x = ap[0];
        a.y = ap[1];
        const float* kp0 = KcT + ((size_t)b * P_ + ka) * L_ + s;
#pragma unroll
        for (int i = 0; i < 12; ++i) {
          const int col = cbase + i * 16 + r;
          v2f bb;
          bb.x = kp0[col];
          bb.y = kp0[L_ + col];
          sa[i] = wmma_f32_4(a, bb, sa[i]);
        }
      }
#pragma unroll
      for (int i = 0; i < 12; ++i) {
        const int col = cbase + i * 16 + r;
#pragma unroll
        for (int j = 0; j < 8; ++j)
          s_alpha[(j + 8 * kh) * APAD_ + col] = sa[i][j] * 0.0625f; // P^-0.5
      }
    }
    __syncthreads();

    // ---- stage 2: masked softmax over 1536, 2 rows per wave ----
    {
      const unsigned char* mrow = mask + (size_t)b * L_ + s;
#pragma unroll
      for (int rr = 2 * wv; rr < 2 * wv + 2; ++rr) {
        float mx = -INFINITY;
        for (int c = lane; c < WIN_; c += 32) {
          float v = s_alpha[rr * APAD_ + c];
          v = mrow[c] ? v : -1e30f;
          mx = fmaxf(mx, v);
        }
        mx = wave_max32(mx);
        float sm = 0.f;
        for (int c = lane; c < WIN_; c += 32) {
          float v = s_alpha[rr * APAD_ + c];
          float e = mrow[c] ? __expf(v - mx) : 0.f;
          sm += e;
          s_alpha[rr * APAD_ + c] = e;
        }
        sm = wave_sum32(sm);
        const float inv = 1.f / sm;
        for (int c = lane; c < WIN_; c += 32) s_alpha[rr * APAD_ + c] *= inv;
      }
    }
    __syncthreads();

    // ---- stage 3: Zk = alpha @ H_win ; 8 col-tiles per wave ----
    v8f zk[8];
#pragma unroll
    for (int ct = 0; ct < 8; ++ct) zk[ct] = (v8f){0.f,0.f,0.f,0.f,0.f,0.f,0.f,0.f};
    for (int k0 = 0; k0 < WIN_; k0 += 4) {
      const int ka = k0 + 2 * kh;
      v2f a;
      a.x = s_alpha[r * APAD_ + ka];        // conflict-free: APAD_%64==4
      a.y = s_alpha[r * APAD_ + ka + 1];
      const float* hp = H + ((size_t)b * L_ + s + ka) * DH_;
#pragma unroll
      for (int ct = 0; ct < 8; ++ct) {
        const int col = col0w + ct * 16 + r;
        v2f bb;
        bb.x = hp[col];
        bb.y = hp[DH_ + col];
        zk[ct] = wmma_f32_4(a, bb, zk[ct]);
      }
    }

    // ---- stage 4: lw partials (deterministic, no atomics) ----
#pragma unroll
    for (int j = 0; j < 8; ++j) {
      const int rr = j + 8 * kh;
      const float* up = u + (size_t)(rowQ + rr) * DH_;
      float p = 0.f;
#pragma unroll
      for (int ct = 0; ct < 8; ++ct) p += zk[ct][j] * up[col0w + ct * 16 + r];
      s_part[rr][wv * 16 + r] = p;
    }
    __syncthreads();

    // ---- stage 5: reduce lw, online-softmax scalars (2 rows per wave) ----
#pragma unroll
    for (int rr = 2 * wv; rr < 2 * wv + 2; ++rr) {
      float p = 0.f;
      for (int c = lane; c < 128; c += 32) p += s_part[rr][c];
      p = wave_sum32(p);
      if (lane == 0) {
        const float lw    = p * 0.03125f;            // D^-0.5, D=1024
        const float mo    = s_m[rr];
        const float mn    = fmaxf(mo, lw);
        const float em    = __expf(mo - mn);
        const float ew    = __expf(lw - mn);
        s_sum[rr] = s_sum[rr] * em + ew;
        s_m[rr]   = mn;
        s_em[rr]  = em;
        s_ew[rr]  = ew;
      }
    }
    __syncthreads();

    // ---- stage 6: z = z*em + zk*ew ----
#pragma unroll
    for (int j = 0; j < 8; ++j) {
      const int rr = j + 8 * kh;
      const float em = s_em[rr], ew = s_ew[rr];
#pragma unroll
      for (int ct = 0; ct < 8; ++ct) z[ct][j] = z[ct][j] * em + zk[ct][j] * ew;
    }
    __syncthreads();
  }

  // ---- epilogue: emit unnormalized group partial (m, ssum, z) ----
  float* zo = zpart + (size_t)g * RT_ * DH_;
#pragma unroll
  for (int j = 0; j < 8; ++j) {
    const int rr = j + 8 * kh;
#pragma unroll
    for (int ct = 0; ct < 8; ++ct) {
      const int col = col0w + ct * 16 + r;
      zo[(size_t)(rowQ + rr) * DH_ + col] = z[ct][j];
    }
  }
  if (tid < 16) {
    mpart[(size_t)g * RT_ + rowQ + tid] = s_m[tid];
    spart[(size_t)g * RT_ + rowQ + tid] = s_sum[tid];
  }
}

// ---------------------------------------------------------------------------
// Exact flash-combine of the NG_=2 window-group partials + final normalize:
//   m = max(m0,m1); e_g = exp(m_g - m)
//   out = (z0*e0 + z1*e1) / (s0*e0 + s1*e1 + 1e-8)
// ---------------------------------------------------------------------------
__global__ void merge_partials(const float* __restrict__ zpart, // [NG,RT,1024]
                               const float* __restrict__ mpart, // [NG,RT]
                               const float* __restrict__ spart, // [NG,RT]
                               float* __restrict__ out) {       // [RT,1024]
  const size_t idx = (size_t)blockIdx.x * blockDim.x + threadIdx.x;
  const size_t row = idx >> 10;                 // / DH_
  const float  m0  = mpart[row];
  const float  m1  = mpart[RT_ + row];
  const float  mn  = fmaxf(m0, m1);
  const float  e0  = __expf(m0 - mn);
  const float  e1  = __expf(m1 - mn);
  const float  den = spart[row] * e0 + spart[RT_ + row] * e1 + 1e-8f;
  out[idx] = (zpart[idx] * e0 + zpart[(size_t)RT_ * DH_ + idx] * e1) / den;
}

// ---------------------------------------------------------------------------
extern "C" void kernel_launch(void* const* d_in, const int* in_sizes, int n_in,
                              void* d_out, int out_size, void* d_ws,
                              size_t ws_size, hipStream_t stream) {
  const float* H       = (const float*)d_in[0];          // [4,8192,1024]
  const float* G       = (const float*)d_in[1];          // [4,512,256]
  const unsigned char* mask = (const unsigned char*)d_in[2]; // [4,8192] bool
  const float* Wq_core = (const float*)d_in[3];          // [256,256]
  const float* Wk_core = (const float*)d_in[4];          // [1024,256]
  const float* Wq_win  = (const float*)d_in[5];          // [256,1024]
  const float* Wk_win  = (const float*)d_in[6];          // [1024,1024]
  float* out = (float*)d_out;                            // [4,512,1024]

  // workspace (floats): qc | Wcomb | u | KcT | zpart | mpart | spart  (~59 MB)
  float* qc    = (float*)d_ws;                           // 2048*256
  float* Wcomb = qc + (size_t)RT_ * P_;                  // 256*1024
  float* u     = Wcomb + (size_t)DG_ * DH_;              // 2048*1024
  float* KcT   = u + (size_t)RT_ * DH_;                  // 4*256*8192
  float* zpart = KcT + (size_t)B_ * P_ * L_;             // NG*2048*1024
  float* mpart = zpart + (size_t)NG_ * RT_ * DH_;        // NG*2048
  float* spart = mpart + (size_t)NG_ * RT_;              // NG*2048

  dim3 blk(256);

  // 1) qc = G @ Wq_core           [2048,256] = [2048,256]@[256,256]
  gemm_wmma_f32<false><<<dim3(RT_ / 32, P_ / 256), blk, 0, stream>>>(
      G, Wq_core, qc, RT_, DG_, P_);

  // 2) Wcomb = Wq_win @ Wk_win^T  [256,1024] = [256,1024]@[1024,1024]^T
  gemm_wmma_f32<true><<<dim3(DG_ / 32, DH_ / 256), blk, 0, stream>>>(
      Wq_win, Wk_win, Wcomb, DG_, DH_, DH_);

  // 3) u = G @ Wcomb              [2048,1024] = [2048,256]@[256,1024]
  gemm_wmma_f32<false><<<dim3(RT_ / 32, DH_ / 256), blk, 0, stream>>>(
      G, Wcomb, u, RT_, DG_, DH_);

  // 4) KcT[b][p][l] = (H @ Wk_core)^T  (per batch)
  kproj_wmma_f32<<<dim3((B_ * L_) / 32, 1), blk, 0, stream>>>(H, Wk_core, KcT);

  // 5) fused windowed flash-pooling, 2 window groups per (b, t-tile)
  fused_gowatti<<<dim3(B_ * 32, NG_), blk, 0, stream>>>(qc, u, KcT, H, mask,
                                                        zpart, mpart, spart);

  // 6) exact combine + normalize
  merge_partials<<<dim3((RT_ * DH_) / 256), blk, 0, stream>>>(zpart, mpart,
                                                              spart, out);
}